// Top1Router_71571335020916
// MI455X (gfx1250) — compile-verified
//
#include <hip/hip_runtime.h>

typedef __attribute__((ext_vector_type(4))) float v4f;

#define EXPERTS 64
#define MAX_LDS_TOKENS 8192

// ---------------------------------------------------------------------------
// Kernel 1: stream 671MB of zeros with non-temporal b128 stores.
// Output is write-once, never re-read by the GPU -> bypass-friendly NT hint.
// ---------------------------------------------------------------------------
__global__ __launch_bounds__(256)
void zero_fill_kernel(float* __restrict__ out, long n) {
  long n4 = n >> 2;
  long gid = (long)blockIdx.x * blockDim.x + threadIdx.x;
  long stride = (long)gridDim.x * blockDim.x;
  v4f z = {0.0f, 0.0f, 0.0f, 0.0f};
  v4f* __restrict__ o4 = (v4f*)out;
  for (long i = gid; i < n4; i += stride) {
    __builtin_nontemporal_store(z, o4 + i);
  }
  // scalar tail (out_size is divisible by 4 in practice, but stay general)
  long rem_start = n4 << 2;
  for (long i = rem_start + gid; i < n; i += stride) {
    out[i] = 0.0f;
  }
}

// ---------------------------------------------------------------------------
// Kernel 2: per-token router. First-occurrence argmax over 64 experts and
// softmax prob at the argmax: p = 1 / sum(exp(x - max)).
// ---------------------------------------------------------------------------
__global__ __launch_bounds__(256)
void router_kernel(const float* __restrict__ x,
                   int* __restrict__ top1,
                   float* __restrict__ wts,
                   int T) {
  int t = blockIdx.x * blockDim.x + threadIdx.x;
  if (t >= T) return;

  const v4f* __restrict__ row = (const v4f*)(x + (size_t)t * EXPERTS);
  __builtin_prefetch(row, 0, 0);  // global_prefetch_b8

  float m = -3.402823466e38f;
  int bi = 0;
#pragma unroll
  for (int j = 0; j < EXPERTS / 4; ++j) {
    v4f v = row[j];
    if (v.x > m) { m = v.x; bi = 4 * j + 0; }
    if (v.y > m) { m = v.y; bi = 4 * j + 1; }
    if (v.z > m) { m = v.z; bi = 4 * j + 2; }
    if (v.w > m) { m = v.w; bi = 4 * j + 3; }
  }

  float s = 0.0f;
#pragma unroll
  for (int j = 0; j < EXPERTS / 4; ++j) {
    v4f v = row[j];
    s += expf(v.x - m);
    s += expf(v.y - m);
    s += expf(v.z - m);
    s += expf(v.w - m);
  }

  top1[t] = bi;
  wts[t] = 1.0f / s;
}

// ---------------------------------------------------------------------------
// Kernel 3: one block per expert. Stage top1[] (32KB) into LDS via gfx1250
// async global->LDS b128 copies, ordered prefix scan of membership, scatter
// combine weight + mask flag for ranks < capacity.
// ---------------------------------------------------------------------------
__global__ __launch_bounds__(256)
void rank_scatter_kernel(const int* __restrict__ top1,
                         const float* __restrict__ wts,
                         float* __restrict__ out,
                         int T, int cap, long N) {
  __shared__ __align__(16) int s_top1[MAX_LDS_TOKENS];
  __shared__ int s_cnt[256];

  const int e    = blockIdx.x;
  const int tid  = threadIdx.x;
  const int nthr = blockDim.x;

  const int* tp = top1;
  if (T <= MAX_LDS_TOKENS && (T & 3) == 0) {
    // Async DMA-style staging: each lane issues b128 async-to-LDS copies.
    unsigned lds_base = (unsigned)(size_t)(&s_top1[0]);
    unsigned long long gbase = (unsigned long long)(size_t)top1;
    int chunks = (T * 4) >> 4;  // number of 16B chunks
    for (int c = tid; c < chunks; c += nthr) {
      unsigned lds_addr = lds_base + ((unsigned)c << 4);
      unsigned long long gaddr = gbase + ((unsigned long long)c << 4);
      asm volatile("global_load_async_to_lds_b128 %0, %1, off"
                   :: "v"(lds_addr), "v"(gaddr)
                   : "memory");
    }
    asm volatile("s_wait_asynccnt 0x0" ::: "memory");
    __syncthreads();
    tp = s_top1;
  }

  // Contiguous token range per thread preserves the cumsum ordering.
  int tpt = (T + nthr - 1) / nthr;
  int t0 = tid * tpt;
  int t1 = t0 + tpt; if (t1 > T) t1 = T;

  int cnt = 0;
  for (int t = t0; t < t1; ++t) cnt += (tp[t] == e) ? 1 : 0;
  s_cnt[tid] = cnt;
  __syncthreads();

  // Block-level exclusive scan (256 entries; serial scan is ~free here).
  if (tid == 0) {
    int run = 0;
    for (int i = 0; i < nthr; ++i) { int c = s_cnt[i]; s_cnt[i] = run; run += c; }
  }
  __syncthreads();

  int rank = s_cnt[tid];
  for (int t = t0; t < t1; ++t) {
    if (tp[t] == e) {
      if (rank < cap) {
        long idx = ((long)t * EXPERTS + e) * (long)cap + rank;
        float w = wts[t];
        out[idx]     = w;     // combine_weights
        out[N + idx] = 1.0f;  // sec_mask (bool as float)
      }
      ++rank;  // ranks counted over the unfiltered cumsum, as in reference
    }
  }
}

// ---------------------------------------------------------------------------
extern "C" void kernel_launch(void* const* d_in, const int* in_sizes, int n_in,
                              void* d_out, int out_size, void* d_ws, size_t ws_size,
                              hipStream_t stream) {
  const float* x = (const float*)d_in[0];
  const int E = EXPERTS;
  int T = in_sizes[0] / E;

  // capacity = floor(1.25*T/E), rounded up to even, min 4
  int cap = (5 * T) / (4 * E);
  cap += cap & 1;
  if (cap < 4) cap = 4;

  long N = (long)T * E * cap;          // elements per output tensor
  float* out = (float*)d_out;
  int*   top1 = (int*)d_ws;
  float* wts  = (float*)((char*)d_ws + sizeof(int) * (size_t)T);

  // 1) zero both output tensors (bandwidth-bound: ~671MB of NT stores)
  zero_fill_kernel<<<8192, 256, 0, stream>>>(out, (long)out_size);

  // 2) per-token argmax + softmax weight
  router_kernel<<<(T + 255) / 256, 256, 0, stream>>>(x, top1, wts, T);

  // 3) per-expert ordered rank + sparse scatter
  rank_scatter_kernel<<<E, 256, 0, stream>>>(top1, wts, out, T, cap, N);
}